// SampledSoftMaxCrossEntropy_23098334117905
// MI455X (gfx1250) — compile-verified
//
#include <hip/hip_runtime.h>

// CDNA5 / gfx1250, wave32.
typedef __attribute__((ext_vector_type(2))) float v2f;
typedef __attribute__((ext_vector_type(8))) float v8f;

#define B_N 1024
#define S_N 2048
#define E_N 128
#define THREADS 256
#define WAVES 8                       // 256 threads / wave32
#define S_PER_WAVE (S_N / WAVES)      // 256 samples per wave
#define TILES_PER_WAVE (S_PER_WAVE / 16)

// One block per batch row b:
//   - stage inputs[b,0:128] and all 2048 sample ids in LDS (coalesced)
//   - each wave computes 256 logits, 16 samples at a time via a chain of
//     32x V_WMMA_F32_16X16X4_F32 (A = inputs[b] replicated over M,
//     B = 16 gathered weight rows; exact fp32 accumulate, L2-resident)
//   - block-wide logsumexp reduction in LDS
//   - per-row loss (lse - logits[b,0]) written to workspace
__global__ __launch_bounds__(THREADS)
void sampled_ce_rows_kernel(const float* __restrict__ inputs,
                            const float* __restrict__ weight,
                            const float* __restrict__ bias,
                            const int*   __restrict__ sample_ids,
                            float*       __restrict__ row_loss)
{
    __shared__ float s_in[E_N];        // inputs[b,:]
    __shared__ int   s_ids[S_N];       // this row's 2048 sample ids
    __shared__ float s_logits[S_N];    // all 2048 logits of this row
    __shared__ float s_red[THREADS];   // reduction scratch

    const int b    = blockIdx.x;
    const int tid  = threadIdx.x;
    const int lane = tid & 31;
    const int wave = tid >> 5;

    if (tid < E_N) s_in[tid] = inputs[b * E_N + tid];

    // Stage ids: 8 contiguous ints per thread -> 2x global_load_b128 each.
    {
        const int4* __restrict__ src =
            (const int4*)(sample_ids + (long long)b * S_N) + tid * 2;
        int4* dst = (int4*)s_ids + tid * 2;
        dst[0] = src[0];
        dst[1] = src[1];
    }
    __syncthreads();

    // A/B operand slotting for V_WMMA_F32_16X16X4_F32 (ISA 7.12.2):
    //   VGPR0: K=k+0 (lanes 0-15) / K=k+2 (lanes 16-31); VGPR1: K=k+1 / K=k+3
    const int koff = (lane >> 4) << 1;   // 0 or 2
    const int n    = lane & 15;          // sample-in-tile (B-matrix column)

    for (int t = 0; t < TILES_PER_WAVE; ++t) {
        const int s0  = wave * S_PER_WAVE + t * 16;
        const int row = s_ids[s0 + n];                     // ds_load_b32 (fast)
        const float* __restrict__ wr = weight + (long long)row * E_N;
        const float bv = bias[row];                        // overlaps wmma chain

        // Warm the next tile's gathered row while this chain runs.
        if (t + 1 < TILES_PER_WAVE) {
            const int nrow = s_ids[s0 + 16 + n];
            __builtin_prefetch(weight + (long long)nrow * E_N + koff * 16, 0, 1);
        }

        v8f c = {};
        #pragma unroll
        for (int k = 0; k < E_N; k += 4) {
            v2f a, bm;
            a.x  = s_in[k + koff];         // hoisted to VGPRs by compiler
            a.y  = s_in[k + koff + 1];
            bm.x = wr[k + koff];           // global_load_b64, L2-resident gather
            bm.y = wr[k + koff + 1];
            // exact fp32: D = A(16x4) * B(4x16) + C ; all 16 A-rows identical
            c = __builtin_amdgcn_wmma_f32_16x16x4_f32(
                    false, a, false, bm, (short)0, c, false, false);
        }
        // C/D layout: VGPR0, lane l<16 -> M=0, N=l  (all M rows equal)
        const float logit = c[0] + bv;
        if (lane < 16) s_logits[s0 + lane] = logit;
    }
    __syncthreads();

    // ---- logsumexp over S_N values, fixed-order tree (deterministic) ----
    float m = -3.402823466e38f;
    #pragma unroll
    for (int i = 0; i < S_N / THREADS; ++i)
        m = fmaxf(m, s_logits[tid + i * THREADS]);
    s_red[tid] = m;
    __syncthreads();
    for (int off = THREADS / 2; off > 0; off >>= 1) {
        if (tid < off) s_red[tid] = fmaxf(s_red[tid], s_red[tid + off]);
        __syncthreads();
    }
    const float mx = s_red[0];
    __syncthreads();

    float sum = 0.f;
    #pragma unroll
    for (int i = 0; i < S_N / THREADS; ++i)
        sum += __expf(s_logits[tid + i * THREADS] - mx);
    s_red[tid] = sum;
    __syncthreads();
    for (int off = THREADS / 2; off > 0; off >>= 1) {
        if (tid < off) s_red[tid] += s_red[tid + off];
        __syncthreads();
    }

    if (tid == 0)
        row_loss[b] = mx + __logf(s_red[0]) - s_logits[0];
}

// Deterministic final mean over the 1024 per-row losses.
__global__ __launch_bounds__(THREADS)
void sampled_ce_mean_kernel(const float* __restrict__ row_loss,
                            float*       __restrict__ out)
{
    __shared__ float s[THREADS];
    const int tid = threadIdx.x;
    s[tid] = row_loss[tid] + row_loss[tid + 256] +
             row_loss[tid + 512] + row_loss[tid + 768];
    __syncthreads();
    for (int off = THREADS / 2; off > 0; off >>= 1) {
        if (tid < off) s[tid] += s[tid + off];
        __syncthreads();
    }
    if (tid == 0) out[0] = s[0] * (1.0f / (float)B_N);
}

extern "C" void kernel_launch(void* const* d_in, const int* in_sizes, int n_in,
                              void* d_out, int out_size, void* d_ws, size_t ws_size,
                              hipStream_t stream)
{
    const float* inputs     = (const float*)d_in[0];   // [B,E]
    const float* weight     = (const float*)d_in[1];   // [V,E]
    const float* bias       = (const float*)d_in[2];   // [V]
    const int*   sample_ids = (const int*)  d_in[3];   // [B,S]
    float* out = (float*)d_out;                        // scalar loss
    float* ws  = (float*)d_ws;                         // B_N floats of scratch

    sampled_ce_rows_kernel<<<B_N, THREADS, 0, stream>>>(inputs, weight, bias,
                                                        sample_ids, ws);
    sampled_ce_mean_kernel<<<1, THREADS, 0, stream>>>(ws, out);
}